// BERTTrainer_4947802325049
// MI455X (gfx1250) — compile-verified
//
#include <hip/hip_runtime.h>
#include <math.h>

typedef __attribute__((ext_vector_type(2))) float v2f;
typedef __attribute__((ext_vector_type(8))) float v8f;

#define L_MAX 16
#define MARGIN 0.4f
#define EPSV 1e-8f
#define D_DIM 1024
#define S_DIM 512
#define CHUNK 128                 // dims per LDS-staged chunk
#define NCHUNK (D_DIM / CHUNK)    // 8
#define PAIRS_PER_BLOCK 16
#define ROWSTRIDE 132             // padded LDS row stride (floats): 132%64=4 -> conflict-free column reads
#define BLOCK_THREADS 512         // 16 waves of 32

__global__ void span_loss_zero_kernel(float* out) {
    if (threadIdx.x == 0) out[0] = 0.0f;
}

__global__ __launch_bounds__(BLOCK_THREADS)
void span_cos_loss_kernel(const float* __restrict__ emb,
                          const int* __restrict__ sid,
                          const int* __restrict__ s_start,
                          const int* __restrict__ s_len,
                          const int* __restrict__ t_start,
                          const int* __restrict__ t_len,
                          const int* __restrict__ labels,
                          float* __restrict__ out, int N) {
    __shared__ float ldsS[PAIRS_PER_BLOCK * ROWSTRIDE];
    __shared__ float ldsT[PAIRS_PER_BLOCK * ROWSTRIDE];
    __shared__ float cfull[3][8][32];   // full 16x16 f32 accumulators for the 3 Gram products

    const int tid  = threadIdx.x;
    const int wave = tid >> 5;          // 0..15, uniform per wave
    const int lane = tid & 31;
    const int pair0 = blockIdx.x * PAIRS_PER_BLOCK;

    // Wave w pools pair pair0+w (clamped for ragged tail; tail lanes masked at the end).
    int myPair = pair0 + wave;
    if (myPair >= N) myPair = N - 1;
    const int    mySid = sid[myPair];
    const int    myS0  = s_start[myPair], mySL = s_len[myPair];
    const int    myT0  = t_start[myPair], myTL = t_len[myPair];
    const size_t sentBase = (size_t)mySid * (size_t)S_DIM * (size_t)D_DIM;

    v8f c_acc = {};                     // per-wave Gram accumulator (waves 0..2 only)

    const float* sRowLds = &ldsS[wave * ROWSTRIDE + (lane << 2)];
    (void)sRowLds;

    for (int c = 0; c < NCHUNK; ++c) {
        const int dim0 = c * CHUNK + (lane << 2);

        // ---- span max-pool (s span) : one float4 per lane per row, coalesced 512B/wave ----
        {
            const size_t rowBase = sentBase + (size_t)myS0 * D_DIM + dim0;
            float4 m = *(const float4*)(emb + rowBase);
            for (int r = 1; r <= mySL; ++r) {
                float4 v = *(const float4*)(emb + rowBase + (size_t)r * D_DIM);
                m.x = fmaxf(m.x, v.x); m.y = fmaxf(m.y, v.y);
                m.z = fmaxf(m.z, v.z); m.w = fmaxf(m.w, v.w);
            }
            float* p = &ldsS[wave * ROWSTRIDE + (lane << 2)];
            p[0] = m.x; p[1] = m.y; p[2] = m.z; p[3] = m.w;
        }
        // ---- span max-pool (t span) ----
        {
            const size_t rowBase = sentBase + (size_t)myT0 * D_DIM + dim0;
            float4 m = *(const float4*)(emb + rowBase);
            for (int r = 1; r <= myTL; ++r) {
                float4 v = *(const float4*)(emb + rowBase + (size_t)r * D_DIM);
                m.x = fmaxf(m.x, v.x); m.y = fmaxf(m.y, v.y);
                m.z = fmaxf(m.z, v.z); m.w = fmaxf(m.w, v.w);
            }
            float* p = &ldsT[wave * ROWSTRIDE + (lane << 2)];
            p[0] = m.x; p[1] = m.y; p[2] = m.z; p[3] = m.w;
        }
        __syncthreads();

        // ---- Gram accumulation via V_WMMA_F32_16X16X4_F32 (waves 0..2; wave-uniform branch,
        //      EXEC stays all-ones as WMMA requires) ----
        if (wave < 3) {
            // wave 0: S x T^T (dot);  wave 1: S x S^T (|s|^2);  wave 2: T x T^T (|t|^2)
            const float* Abase = (wave == 2) ? ldsT : ldsS;
            const float* Bbase = (wave == 1) ? ldsS : ldsT;
            // f32 16x16x4 layout: lane m(0..15) -> K offsets {0,1}; lane m+16 -> {2,3}.
            // B (4x16, N on lanes) indexes LDS identically because column N of B is pair N's vector.
            const int row  = lane & 15;
            const int koff = (lane < 16) ? 0 : 2;
            const float* aRow = Abase + row * ROWSTRIDE + koff;
            const float* bRow = Bbase + row * ROWSTRIDE + koff;
#pragma unroll
            for (int k = 0; k < CHUNK; k += 4) {
                v2f a = *(const v2f*)(aRow + k);   // 8B-aligned ds_load_b64
                v2f b = *(const v2f*)(bRow + k);
                c_acc = __builtin_amdgcn_wmma_f32_16x16x4_f32(
                    false, a, false, b, (short)0, c_acc, false, false);
            }
        }
        __syncthreads();   // before pooling overwrites LDS next chunk
    }

    // ---- spill the three 16x16 accumulators so wave 0 can read the diagonals ----
    if (wave < 3) {
#pragma unroll
        for (int r = 0; r < 8; ++r) cfull[wave][r][lane] = c_acc[r];
    }
    __syncthreads();

    if (wave == 0 && lane < PAIRS_PER_BLOCK) {
        float per = 0.0f;
        const int i = lane;                      // pair index within block
        const bool live = (pair0 + i) < N;
        if (live) {
            // C/D diag: i<8 -> (vgpr i, lane i); i>=8 -> (vgpr i-8, lane i+16)
            float dot, ss, tt;
            if (i < 8) {
                dot = cfull[0][i][i];  ss = cfull[1][i][i];  tt = cfull[2][i][i];
            } else {
                dot = cfull[0][i - 8][i + 16];
                ss  = cfull[1][i - 8][i + 16];
                tt  = cfull[2][i - 8][i + 16];
            }
            const float cosv = dot / (sqrtf(ss) * sqrtf(tt) + EPSV);
            const float y = 2.0f * (float)labels[pair0 + i] - 1.0f;
            per = (y > 0.0f) ? (1.0f - cosv) : fmaxf(0.0f, cosv - MARGIN);
        }
        // reduce 16 lanes -> lane 0 (width 16 keeps sources inside active lanes)
        per += __shfl_down(per, 8, 16);
        per += __shfl_down(per, 4, 16);
        per += __shfl_down(per, 2, 16);
        per += __shfl_down(per, 1, 16);
        if (lane == 0) atomicAdd(out, per / (float)N);
    }
}

extern "C" void kernel_launch(void* const* d_in, const int* in_sizes, int n_in,
                              void* d_out, int out_size, void* d_ws, size_t ws_size,
                              hipStream_t stream) {
    const float* emb     = (const float*)d_in[0];
    const int*   sid     = (const int*)d_in[1];
    const int*   s_start = (const int*)d_in[2];
    const int*   s_len   = (const int*)d_in[3];
    const int*   t_start = (const int*)d_in[4];
    const int*   t_len   = (const int*)d_in[5];
    const int*   labels  = (const int*)d_in[6];
    float*       out     = (float*)d_out;
    const int N = in_sizes[1];   // number of pairs

    span_loss_zero_kernel<<<1, 32, 0, stream>>>(out);
    const int blocks = (N + PAIRS_PER_BLOCK - 1) / PAIRS_PER_BLOCK;
    span_cos_loss_kernel<<<blocks, BLOCK_THREADS, 0, stream>>>(
        emb, sid, s_start, s_len, t_start, t_len, labels, out, N);
}